// HeadQK_71940702208377
// MI455X (gfx1250) — compile-verified
//
#include <hip/hip_runtime.h>

// Problem constants (match reference)
#define TDIM  2048
#define CDIM  1024
#define HQK   256
#define VOCAB 50257

typedef float v2f __attribute__((ext_vector_type(2)));
typedef float v8f __attribute__((ext_vector_type(8)));

// ---------------------------------------------------------------------------
// Zero the [T, VOCAB] output (412 MB) with wide stores. This is the dominant
// HBM cost of the whole problem, so keep it pure streaming float4 stores.
// ---------------------------------------------------------------------------
__global__ void hqk_zero_kernel(float* __restrict__ out, long long n) {
    long long i      = (long long)blockIdx.x * blockDim.x + threadIdx.x;
    long long stride = (long long)gridDim.x * blockDim.x;
    long long n4     = n >> 2;
    float4* o4 = (float4*)out;
    float4  z  = make_float4(0.f, 0.f, 0.f, 0.f);
    for (long long j = i; j < n4; j += stride) o4[j] = z;
    for (long long j = (n4 << 2) + i; j < n; j += stride) out[j] = 0.f;
}

// ---------------------------------------------------------------------------
// Projection GEMM: q = x @ Wq, k = x @ Wk using V_WMMA_F32_16X16X4_F32.
// One wave (32 lanes) owns one 16x16 output tile; K-loop over C in steps of 4.
//
// A-frag (16x4 f32, 2 VGPRs/lane): lanes 0-15 rows M=0..15 hold K={kb,kb+1},
// lanes 16-31 same rows hold K={kb+2,kb+3}  -> one aligned float2 load.
// B-frag (4x16): N = lane&15, K = kb + 2*(lane>>4) + {0,1}.
// C/D (16x16 f32, 8 VGPRs): element (M = r + 8*(lane>>4), N = lane&15).
// ---------------------------------------------------------------------------
__global__ void hqk_proj_kernel(const float* __restrict__ x,
                                const float* __restrict__ Wq,
                                const float* __restrict__ Wk,
                                float* __restrict__ qk) {
    const int rowTile = blockIdx.x;   // 0..T/16-1
    const int colTile = blockIdx.y;   // 0..HQK/16-1
    const int which   = blockIdx.z;   // 0 = q, 1 = k
    const float* W = which ? Wk : Wq;
    float* outp = qk + (long long)which * TDIM * HQK;

    const int lane   = threadIdx.x;
    const int lane16 = lane & 15;
    const int half   = lane >> 4;

    const int   row  = rowTile * 16 + lane16;
    const int   col  = colTile * 16 + lane16;
    const float* xrow = x + (long long)row * CDIM + 2 * half;

    v8f acc = {};
    #pragma unroll 4
    for (int kb = 0; kb < CDIM; kb += 4) {
        v2f a = *(const v2f*)(xrow + kb);
        v2f b;
        b.x = W[(long long)(kb + 2 * half + 0) * HQK + col];
        b.y = W[(long long)(kb + 2 * half + 1) * HQK + col];
        acc = __builtin_amdgcn_wmma_f32_16x16x4_f32(
            /*neg_a=*/false, a, /*neg_b=*/false, b,
            /*c_mod=*/(short)0, acc, /*reuse_a=*/false, /*reuse_b=*/false);
    }

    #pragma unroll
    for (int r = 0; r < 8; ++r) {
        int m = rowTile * 16 + r + 8 * half;
        outp[(long long)m * HQK + colTile * 16 + lane16] = acc[r];
    }
}

// ---------------------------------------------------------------------------
// Score tile + fused causal mask + segment-sum scatter.
// c_tile = q_tile @ k_tileT / HQK ; out[t, idx[s]] += c[t, s] for s <= t.
// kT columns are contiguous k rows -> both fragments come from aligned
// float2 loads. Scatter uses native global_atomic_add_f32 (relaxed/agent).
// ---------------------------------------------------------------------------
__global__ void hqk_scores_scatter_kernel(const float* __restrict__ q,
                                          const float* __restrict__ k,
                                          const int* __restrict__ idx,
                                          float* __restrict__ out) {
    const int sTile = blockIdx.x;   // key tile
    const int tTile = blockIdx.y;   // query tile
    if (sTile > tTile) return;      // wave-uniform: strictly above diagonal

    const int lane   = threadIdx.x;
    const int lane16 = lane & 15;
    const int half   = lane >> 4;

    const float* qrow = q + (long long)(tTile * 16 + lane16) * HQK + 2 * half;
    const float* krow = k + (long long)(sTile * 16 + lane16) * HQK + 2 * half;

    v8f acc = {};
    #pragma unroll 8
    for (int kb = 0; kb < HQK; kb += 4) {
        v2f a = *(const v2f*)(qrow + kb);
        v2f b = *(const v2f*)(krow + kb);
        acc = __builtin_amdgcn_wmma_f32_16x16x4_f32(
            false, a, false, b, (short)0, acc, false, false);
    }

    const int   s     = sTile * 16 + lane16;
    const int   token = idx[s];
    const float scale = 1.0f / (float)HQK;

    #pragma unroll
    for (int r = 0; r < 8; ++r) {
        int t = tTile * 16 + r + 8 * half;
        if (s <= t) {
            float v = acc[r] * scale;
            __hip_atomic_fetch_add(out + (long long)t * VOCAB + token, v,
                                   __ATOMIC_RELAXED, __HIP_MEMORY_SCOPE_AGENT);
        }
    }
}

// ---------------------------------------------------------------------------
// Launch: inputs are x [T*C], idx [T], Wq [C*HQK], Wk [C*HQK]; output is
// [T, VOCAB] fp32. Workspace holds q and k (2 * T * HQK * 4B = 4 MB).
// ---------------------------------------------------------------------------
extern "C" void kernel_launch(void* const* d_in, const int* in_sizes, int n_in,
                              void* d_out, int out_size, void* d_ws, size_t ws_size,
                              hipStream_t stream) {
    const float* x   = (const float*)d_in[0];
    const int*   idx = (const int*)  d_in[1];
    const float* Wq  = (const float*)d_in[2];
    const float* Wk  = (const float*)d_in[3];
    float* out = (float*)d_out;
    float* qbuf = (float*)d_ws;                       // [T, HQK]
    float* kbuf = qbuf + (long long)TDIM * HQK;       // [T, HQK]

    // 1) Zero the 412 MB output (HBM-bound floor of this problem).
    hqk_zero_kernel<<<4096, 256, 0, stream>>>(out, (long long)out_size);

    // 2) q/k projections: 128 x 16 tiles x {q,k}, one wave per 16x16 tile.
    hqk_proj_kernel<<<dim3(TDIM / 16, HQK / 16, 2), 32, 0, stream>>>(x, Wq, Wk,
                                                                     qbuf);

    // 3) Causal score tiles + fused segment-sum scatter into out.
    hqk_scores_scatter_kernel<<<dim3(TDIM / 16, TDIM / 16), 32, 0, stream>>>(
        qbuf, kbuf, idx, out);
}